// CenterNetDecoder_39350490366624
// MI455X (gfx1250) — compile-verified
//
#include <hip/hip_runtime.h>
#include <hip/hip_bf16.h>

// CenterNet decode for MI455X (gfx1250, wave32).
// B=32, C=80, H=W=128, TOPK=100, STRIDE=4, MIN_SCORE=0.05
#define NB    32
#define NC    80
#define NH    128
#define NW    128
#define NHW   (NH * NW)       // 16384
#define NTOPK 100
#define NMAPS (NB * NC)       // 2560
#define NCAND (NC * NTOPK)    // 8000 per batch

typedef unsigned long long u64;
typedef unsigned int       u32;
typedef int v4i __attribute__((vector_size(16)));

// ---- CDNA5 async global->LDS (codegen-confirmed: global_load_async_to_lds_b128)
#if defined(__has_builtin)
#if __has_builtin(__builtin_amdgcn_global_load_async_to_lds_b128)
#define HAVE_ASYNC_B128 1
#endif
#if __has_builtin(__builtin_amdgcn_s_wait_asynccnt)
#define HAVE_WAIT_ASYNC 1
#endif
#endif

__device__ inline u64 wave_max_u64(u64 v) {
  // full-wave (32 lane) max
  #pragma unroll
  for (int m = 16; m > 0; m >>= 1) {
    u64 o = __shfl_xor(v, m, 32);
    if (o > v) v = o;
  }
  return v;
}

// ---------------------------------------------------------------------------
// Kernel A: one block per (batch, class) map.
// async-stage 64KB -> LDS, sigmoid, branchless 3x3 NMS, per-map top-100
// (sorted desc, ties by smaller spatial idx). Emits (score, idx) pairs.
// ---------------------------------------------------------------------------
__global__ __launch_bounds__(512) void kA(const float* __restrict__ hm,
                                          float* __restrict__ cscore,
                                          u32* __restrict__ cidx) {
  __shared__ __align__(16) unsigned char SBUF[NHW * 4];  // exactly 64KB
  float* S    = (float*)SBUF;   // sigmoid scores during NMS phase
  u64*   redA = (u64*)SBUF;     // aliased reduction buffer afterwards

  const int t  = threadIdx.x;          // 0..511
  const int bc = blockIdx.x;           // b*NC + c
  const float* src = hm + (size_t)bc * NHW;

  // ---- stage raw 64KB map into LDS via async DMA (ASYNCcnt-tracked) ----
#if HAVE_ASYNC_B128
  {
    typedef __attribute__((address_space(1))) v4i* gv4_t;
    typedef __attribute__((address_space(3))) v4i* lv4_t;
    gv4_t g = (gv4_t)(const_cast<float*>(src));
    lv4_t l = (lv4_t)S;
    #pragma unroll
    for (int k = 0; k < 8; ++k) {
      int i4 = (k << 9) + t;  // float4 index, 4096 total
      __builtin_amdgcn_global_load_async_to_lds_b128(g + i4, l + i4, 0, 0);
    }
#if HAVE_WAIT_ASYNC
    __builtin_amdgcn_s_wait_asynccnt(0);
#else
    asm volatile("s_wait_asynccnt 0" ::: "memory");
#endif
  }
#else
  #pragma unroll
  for (int k = 0; k < 8; ++k) {
    int i4 = (k << 9) + t;
    ((float4*)S)[i4] = ((const float4*)src)[i4];
  }
#endif
  __syncthreads();

  // ---- in-place sigmoid (each thread owns its strided slots) ----
  #pragma unroll
  for (int k = 0; k < 32; ++k) {
    int i = (k << 9) + t;
    S[i] = 1.0f / (1.0f + expf(-S[i]));
  }
  __syncthreads();

  // ---- branchless 3x3 NMS (clamped coords == -inf padding semantics) ------
  // keep <=> v >= max(3x3 incl. center); clamped coords only duplicate
  // in-range cells, and max over a multiset == max over the set.
  float cand[32];
  #pragma unroll
  for (int k = 0; k < 32; ++k) {
    int i = (k << 9) + t;
    int y = i >> 7, x = i & (NW - 1);
    int ym = (y > 0) ? y - 1 : 0, yp = (y < NH - 1) ? y + 1 : NH - 1;
    int xm = (x > 0) ? x - 1 : 0, xp = (x < NW - 1) ? x + 1 : NW - 1;
    int rm = ym << 7, r0 = y << 7, rp = yp << 7;
    float v = S[r0 + x];
    float m = S[rm + xm];
    m = fmaxf(m, S[rm + x]);
    m = fmaxf(m, S[rm + xp]);
    m = fmaxf(m, S[r0 + xm]);
    m = fmaxf(m, S[r0 + xp]);
    m = fmaxf(m, S[rp + xm]);
    m = fmaxf(m, S[rp + x]);
    m = fmaxf(m, S[rp + xp]);
    cand[k] = (v >= m) ? v : 0.0f;   // suppressed -> 0.0 (matches reference)
  }
  __syncthreads();  // S dead beyond here; redA aliases it

  // ---- iterative top-100: key = (score desc, idx asc), removal = -1 score --
  const int wid = t >> 5;
  const int lane = t & 31;
  #pragma unroll 1   // keep loop rolled
  for (int it = 0; it < NTOPK; ++it) {
    u64 best = 0;
    #pragma unroll
    for (int k = 0; k < 32; ++k) {
      int i = (k << 9) + t;
      float s = cand[k];
      if (s >= 0.0f) {
        u64 key = (((u64)__float_as_uint(s) + 1ull) << 14) | (u32)(0x3FFF - i);
        if (key > best) best = key;
      }
    }
    best = wave_max_u64(best);
    if (lane == 0) redA[wid] = best;   // 16 waves
    __syncthreads();
    if (t < 32) {
      u64 v = (t < 16) ? redA[t] : 0;
      v = wave_max_u64(v);
      if (t == 0) redA[0] = v;
    }
    __syncthreads();
    u64 key = redA[0];
    int idx = 0x3FFF - (int)(key & 0x3FFFull);
    if (t == 0) {
      float s = __uint_as_float((u32)((key >> 14) - 1ull));
      cscore[bc * NTOPK + it] = s;
      cidx[bc * NTOPK + it]   = (u32)idx;
    }
    if ((idx & 511) == t) {            // owner removes the winner
      int kk = idx >> 9;
      #pragma unroll
      for (int k = 0; k < 32; ++k)
        if (k == kk) cand[k] = -1.0f;
    }
    __syncthreads();
  }
}

// ---------------------------------------------------------------------------
// Kernel B: one block per batch. Global top-100 of the 8000 per-class
// candidates (tie-break by array position == reference ordering), then
// gather offsets/wh and emit scores/classes/bboxes.
// ---------------------------------------------------------------------------
__global__ __launch_bounds__(256) void kB(const float* __restrict__ cscore,
                                          const u32* __restrict__ cidx,
                                          const float* __restrict__ offh,
                                          const float* __restrict__ whh,
                                          float* __restrict__ out) {
  __shared__ float sc[NCAND];
  __shared__ u32   si[NCAND];
  __shared__ u64   redB[9];
  const int t = threadIdx.x;
  const int b = blockIdx.x;

  for (int i = t; i < NCAND; i += 256) {
    sc[i] = cscore[b * NCAND + i];
    si[i] = cidx[b * NCAND + i];
  }
  __syncthreads();

  float* out_sc = out;                         // [NB][NTOPK]
  float* out_cl = out + NB * NTOPK;            // [NB][NTOPK]
  float* out_bb = out + 2 * NB * NTOPK;        // [NB][NTOPK][4]
  const float* offb = offh + (size_t)b * 2 * NHW;
  const float* whb  = whh  + (size_t)b * 2 * NHW;
  const int wid = t >> 5, lane = t & 31;

  #pragma unroll 1   // keep loop rolled
  for (int it = 0; it < NTOPK; ++it) {
    u64 best = 0;
    #pragma unroll 1
    for (int i = t; i < NCAND; i += 256) {
      float s = sc[i];
      if (s >= 0.0f) {
        u64 key = (((u64)__float_as_uint(s) + 1ull) << 13) | (u32)(8191 - i);
        if (key > best) best = key;
      }
    }
    best = wave_max_u64(best);
    if (lane == 0) redB[wid] = best;   // 8 waves
    __syncthreads();
    if (t < 32) {
      u64 v = (t < 8) ? redB[t] : 0;
      v = wave_max_u64(v);
      if (t == 0) redB[0] = v;
    }
    __syncthreads();
    if (t == 0) {
      u64 key = redB[0];
      int pos = 8191 - (int)(key & 0x1FFFull);
      float s = sc[pos];
      u32 hw  = si[pos];
      sc[pos] = -1.0f;                  // remove winner
      int cls = pos / NTOPK;
      int y = (int)(hw >> 7), x = (int)(hw & (NW - 1));
      float ox = offb[hw], oy = offb[NHW + hw];
      float bw = whb[hw],  bh = whb[NHW + hw];
      float xs = (float)x + ox, ys = (float)y + oy;
      bool keep = s > 0.05f;            // MIN_SCORE
      out_sc[b * NTOPK + it] = keep ? s : -1.0f;
      out_cl[b * NTOPK + it] = keep ? (float)cls : -1.0f;
      float* bb = out_bb + (size_t)(b * NTOPK + it) * 4;
      bb[0] = keep ? (xs - bw * 0.5f) * 4.0f : 0.0f;  // STRIDE=4
      bb[1] = keep ? (ys - bh * 0.5f) * 4.0f : 0.0f;
      bb[2] = keep ? (xs + bw * 0.5f) * 4.0f : 0.0f;
      bb[3] = keep ? (ys + bh * 0.5f) * 4.0f : 0.0f;
    }
    __syncthreads();
  }
}

extern "C" void kernel_launch(void* const* d_in, const int* in_sizes, int n_in,
                              void* d_out, int out_size, void* d_ws, size_t ws_size,
                              hipStream_t stream) {
  (void)in_sizes; (void)n_in; (void)out_size; (void)ws_size;
  const float* hm   = (const float*)d_in[0];  // heatmap (32,80,128,128)
  const float* offh = (const float*)d_in[1];  // offset  (32,2,128,128)
  const float* whh  = (const float*)d_in[2];  // wh      (32,2,128,128)
  float* out = (float*)d_out;                 // 19200 floats

  float* cscore = (float*)d_ws;                                           // 2560*100 f32
  u32*   cidx   = (u32*)((char*)d_ws + (size_t)NMAPS * NTOPK * sizeof(float));

  kA<<<NMAPS, 512, 0, stream>>>(hm, cscore, cidx);
  kB<<<NB, 256, 0, stream>>>(cscore, cidx, offh, whh, out);
}